// Qwen3Attention_82102594830550
// MI455X (gfx1250) — compile-verified
//
#include <hip/hip_runtime.h>
#include <math.h>

// Problem constants (Qwen3 attention, fixed by the reference)
#define BB    8
#define QQ    16
#define HHID  4096
#define NHH   32
#define KVHH  8
#define DD    128
#define SSEQ  4096
#define GG    4
#define NCHUNK 8
#define CHUNK (SSEQ / NCHUNK)          // 512
#define ATT_SCALE 0.08838834764831845f // D^-0.5
#define EPSN 1e-6f

typedef float v2f __attribute__((ext_vector_type(2)));
typedef float v8f __attribute__((ext_vector_type(8)));

static __device__ __forceinline__ v8f wmma_f32(v2f a, v2f b, v8f c) {
  // D = A(16x4 f32) * B(4x16 f32) + C(16x16 f32)  ->  v_wmma_f32_16x16x4_f32
  return __builtin_amdgcn_wmma_f32_16x16x4_f32(false, a, false, b, (short)0, c,
                                               false, false);
}

// ---------------------------------------------------------------------------
// GEMM: C(128 x N) = A(128 x K) * W(K x N), all row-major fp32.
// One workgroup per 16-wide N tile; 8 waves, wave w owns M-tile rows 16w..16w+15.
// Weight slab (128 x 16 per K-chunk) staged through LDS so each weight element
// is fetched from HBM exactly once; A rows are L2-resident (<=2MB).
// ---------------------------------------------------------------------------
__global__ __launch_bounds__(256) void gemm_rows128_kernel(
    const float* __restrict__ A, const float* __restrict__ W,
    float* __restrict__ C, int K, int N) {
  __shared__ __align__(16) float sW[128 * 16];
  const int tid  = threadIdx.x;
  const int wave = tid >> 5;
  const int lane = tid & 31;
  const int lm   = lane & 15;
  const int lh   = lane >> 4;          // 0 or 1 (lane half)
  const int col0 = blockIdx.x * 16;
  const int m0   = wave * 16;
  const float* Arow = A + (size_t)(m0 + lm) * K;

  v8f acc = {0, 0, 0, 0, 0, 0, 0, 0};

  for (int kc = 0; kc < K; kc += 128) {
    // cooperative stage of W[kc..kc+127][col0..col0+15] (2048 floats, 512 x float4)
    for (int i = tid; i < 512; i += 256) {
      const int r  = i >> 2;
      const int c4 = i & 3;
      *(float4*)(sW + r * 16 + c4 * 4) =
          *(const float4*)(W + (size_t)(kc + r) * N + col0 + c4 * 4);
    }
    __syncthreads();
#pragma unroll
    for (int t = 0; t < 32; ++t) {
      const int k = t * 4;
      // A frag: VGPR0 = K=k+2lh, VGPR1 = K=k+1+2lh (contiguous -> b64 load)
      v2f a = *(const v2f*)(Arow + kc + k + 2 * lh);
      v2f b;
      b.x = sW[(k + 2 * lh) * 16 + lm];
      b.y = sW[(k + 1 + 2 * lh) * 16 + lm];
      acc = wmma_f32(a, b, acc);
    }
    __syncthreads();
  }
  // C/D layout: VGPR v -> row (v + 8*lh), col lane%16
#pragma unroll
  for (int v = 0; v < 8; ++v)
    C[(size_t)(m0 + v + 8 * lh) * N + col0 + lm] = acc[v];
}

// ---------------------------------------------------------------------------
// Fused RMSNorm + RoPE, in place on (B, Q, heads, 128) rows. One wave per row.
// ---------------------------------------------------------------------------
__global__ __launch_bounds__(256) void rmsnorm_rope_kernel(
    float* __restrict__ X, const float* __restrict__ nw,
    const float* __restrict__ cosb, const float* __restrict__ sinb,
    int heads, int nrows) {
  const int wave = threadIdx.x >> 5;
  const int lane = threadIdx.x & 31;
  const int row  = blockIdx.x * 8 + wave;
  if (row >= nrows) return;
  const int b   = row / (QQ * heads);
  const int rem = row % (QQ * heads);
  const int q   = rem / heads;

  float* x = X + (size_t)row * DD;
  const int d0 = lane;  // handles d0, d0+32, d0+64, d0+96
  float x0 = x[d0], x1 = x[d0 + 32], x2 = x[d0 + 64], x3 = x[d0 + 96];
  float ss = x0 * x0 + x1 * x1 + x2 * x2 + x3 * x3;
#pragma unroll
  for (int off = 16; off > 0; off >>= 1) ss += __shfl_xor(ss, off, 32);
  const float inv = rsqrtf(ss * (1.0f / DD) + EPSN);
  x0 *= inv * nw[d0];      x1 *= inv * nw[d0 + 32];
  x2 *= inv * nw[d0 + 64]; x3 *= inv * nw[d0 + 96];

  const float* cr = cosb + ((size_t)b * QQ + q) * DD;
  const float* sr = sinb + ((size_t)b * QQ + q) * DD;
  const float c0 = cr[d0], c1 = cr[d0 + 32], c2 = cr[d0 + 64], c3 = cr[d0 + 96];
  const float s0 = sr[d0], s1 = sr[d0 + 32], s2 = sr[d0 + 64], s3 = sr[d0 + 96];
  // rotate_half: out[d<64] = x[d]*c - x[d+64]*s ; out[d>=64] = x[d]*c + x[d-64]*s
  x[d0]      = x0 * c0 - x2 * s0;
  x[d0 + 32] = x1 * c1 - x3 * s1;
  x[d0 + 64] = x2 * c2 + x0 * s2;
  x[d0 + 96] = x3 * c3 + x1 * s3;
}

// ---------------------------------------------------------------------------
// Inverse scatter map: map[b*S + s] = source flat index (b_src*Q + q_src) or -1.
// Scatter targets (b,s) are unique in the reference, so this is race-free.
// ---------------------------------------------------------------------------
__global__ void map_init_kernel(int* __restrict__ map, int n) {
  const int i = blockIdx.x * blockDim.x + threadIdx.x;
  if (i < n) map[i] = -1;
}

__global__ void map_scatter_kernel(int* __restrict__ map,
                                   const int* __restrict__ cbi,
                                   const int* __restrict__ csi, int n) {
  const int i = blockIdx.x * blockDim.x + threadIdx.x;
  if (i < n) {
    const int tb = cbi[i];
    const int ts = csi[i];
    if (tb >= 0 && tb < BB && ts >= 0 && ts < SSEQ) map[tb * SSEQ + ts] = i;
  }
}

// ---------------------------------------------------------------------------
// Flash-attention partial: one wave per (b, kv, g, chunk-of-512-keys).
// Scores and P*V both via v_wmma_f32_16x16x4_f32. K/V rows are patched on
// read through `map` (new K/V from this step vs. original cache) and staged
// into LDS; P is transposed C-layout -> A-layout through LDS.
// ---------------------------------------------------------------------------
__global__ __launch_bounds__(32) void attn_partial_kernel(
    const float* __restrict__ qn, const float* __restrict__ kn,
    const float* __restrict__ vn, const float* __restrict__ kcache,
    const float* __restrict__ vcache, const float* __restrict__ mask,
    const int* __restrict__ map, float* __restrict__ partO,
    float* __restrict__ partM, float* __restrict__ partL) {
  __shared__ __align__(16) float tile[16 * 128];  // staged K tile, then V tile
  __shared__ __align__(16) float pbuf[16 * 16];   // P transpose bounce buffer

  int x = blockIdx.x;
  const int chunk = x & (NCHUNK - 1); x >>= 3;
  const int g     = x & (GG - 1);     x >>= 2;
  const int kv    = x & (KVHH - 1);   x >>= 3;
  const int b     = x;
  const int h     = kv * GG + g;
  const int lane  = threadIdx.x;
  const int lm    = lane & 15;
  const int lh    = lane >> 4;

  // Preload Q A-fragments for this head's 16 query rows (reused for all tiles)
  v2f qa[32];
  const float* qrow = qn + (((size_t)b * QQ + lm) * NHH + h) * DD;
#pragma unroll
  for (int t = 0; t < 32; ++t) qa[t] = *(const v2f*)(qrow + t * 4 + 2 * lh);

  v8f o[8];
  float mrow[8], lrow[8];
#pragma unroll
  for (int n = 0; n < 8; ++n) o[n] = (v8f){0, 0, 0, 0, 0, 0, 0, 0};
#pragma unroll
  for (int v = 0; v < 8; ++v) { mrow[v] = -3.0e38f; lrow[v] = 0.0f; }

  const int s0 = chunk * CHUNK;
  for (int st = s0; st < s0 + CHUNK; st += 16) {
    const int s   = st + lm;                 // lane lm owns key row s of the tile
    const int src = map[b * SSEQ + s];
    // ---- stage (patched) K tile [16 s][128 d] into LDS
    const float* krow =
        (src >= 0) ? (kn + ((size_t)src * KVHH + kv) * DD)
                   : (kcache + (((size_t)b * KVHH + kv) * SSEQ + s) * DD);
#pragma unroll
    for (int i = 0; i < 16; ++i)
      *(float4*)(tile + lm * 128 + lh * 64 + i * 4) =
          *(const float4*)(krow + lh * 64 + i * 4);
    __syncthreads();
    // ---- scores tile (16q x 16s) = Q * K^T, contraction over D=128
    v8f c = {0, 0, 0, 0, 0, 0, 0, 0};
#pragma unroll
    for (int t = 0; t < 32; ++t) {
      v2f bb = *(const v2f*)(tile + lm * 128 + t * 4 + 2 * lh);
      c = wmma_f32(qa[t], bb, c);
    }
    __syncthreads();
    // ---- scale + mask + online softmax (rows live in 16-lane halves)
#pragma unroll
    for (int v = 0; v < 8; ++v) {
      const int qrow_idx = v + 8 * lh;
      float sc = c[v] * ATT_SCALE + mask[((size_t)b * QQ + qrow_idx) * SSEQ + s];
      float cm = sc;
      cm = fmaxf(cm, __shfl_xor(cm, 8, 16));
      cm = fmaxf(cm, __shfl_xor(cm, 4, 16));
      cm = fmaxf(cm, __shfl_xor(cm, 2, 16));
      cm = fmaxf(cm, __shfl_xor(cm, 1, 16));
      const float mnew  = fmaxf(mrow[v], cm);
      const float p     = __expf(sc - mnew);
      const float alpha = __expf(mrow[v] - mnew);
      float ps = p;
      ps += __shfl_xor(ps, 8, 16);
      ps += __shfl_xor(ps, 4, 16);
      ps += __shfl_xor(ps, 2, 16);
      ps += __shfl_xor(ps, 1, 16);
      lrow[v] = lrow[v] * alpha + ps;
      mrow[v] = mnew;
#pragma unroll
      for (int n = 0; n < 8; ++n) o[n][v] *= alpha;
      pbuf[qrow_idx * 16 + lm] = p;  // C layout -> LDS for transpose
    }
    __syncthreads();
    // ---- reload P as A-fragments (K dim = 16 keys -> 4 wmma steps)
    v2f pa[4];
#pragma unroll
    for (int t = 0; t < 4; ++t)
      pa[t] = *(const v2f*)(pbuf + lm * 16 + t * 4 + 2 * lh);
    __syncthreads();
    // ---- stage (patched) V tile, reusing `tile`
    const float* vrow =
        (src >= 0) ? (vn + ((size_t)src * KVHH + kv) * DD)
                   : (vcache + (((size_t)b * KVHH + kv) * SSEQ + s) * DD);
#pragma unroll
    for (int i = 0; i < 16; ++i)
      *(float4*)(tile + lm * 128 + lh * 64 + i * 4) =
          *(const float4*)(vrow + lh * 64 + i * 4);
    __syncthreads();
    // ---- O += P * V  (8 d-tiles of 16)
#pragma unroll
    for (int n = 0; n < 8; ++n) {
      v8f oc = o[n];
#pragma unroll
      for (int t = 0; t < 4; ++t) {
        v2f bb;
        bb.x = tile[(t * 4 + 2 * lh) * 128 + n * 16 + lm];
        bb.y = tile[(t * 4 + 1 + 2 * lh) * 128 + n * 16 + lm];
        oc = wmma_f32(pa[t], bb, oc);
      }
      o[n] = oc;
    }
    __syncthreads();
  }
  // ---- write partial O (16x128), m, l
  float* Od = partO + (size_t)blockIdx.x * (QQ * DD);
#pragma unroll
  for (int n = 0; n < 8; ++n)
#pragma unroll
    for (int v = 0; v < 8; ++v)
      Od[(v + 8 * lh) * DD + n * 16 + lm] = o[n][v];
  if (lm == 0) {
#pragma unroll
    for (int v = 0; v < 8; ++v) {
      partM[(size_t)blockIdx.x * QQ + v + 8 * lh] = mrow[v];
      partL[(size_t)blockIdx.x * QQ + v + 8 * lh] = lrow[v];
    }
  }
}

// ---------------------------------------------------------------------------
// Combine the NCHUNK softmax partials per (b, h, q) row; write ctx (B,Q,NH*D).
// ---------------------------------------------------------------------------
__global__ __launch_bounds__(128) void attn_combine_kernel(
    const float* __restrict__ partO, const float* __restrict__ partM,
    const float* __restrict__ partL, float* __restrict__ ctx) {
  const int d = threadIdx.x;
  int x = blockIdx.x;  // = (b*NH + h)*Q + q
  const int q = x % QQ;  x /= QQ;
  const int h = x % NHH; x /= NHH;
  const int b = x;
  const int kv = h / GG, g = h % GG;
  const size_t p0 = (((size_t)b * KVHH + kv) * GG + g) * NCHUNK;

  float M = -3.0e38f;
#pragma unroll
  for (int c = 0; c < NCHUNK; ++c) M = fmaxf(M, partM[(p0 + c) * QQ + q]);
  float L = 0.0f, acc = 0.0f;
#pragma unroll
  for (int c = 0; c < NCHUNK; ++c) {
    const float w = __expf(partM[(p0 + c) * QQ + q] - M);
    L += partL[(p0 + c) * QQ + q] * w;
    acc += partO[((p0 + c) * QQ + q) * DD + d] * w;
  }
  ctx[(((size_t)b * QQ + q) * NHH + h) * DD + d] = acc / L;
}

// ---------------------------------------------------------------------------
// Host-side orchestration
// ---------------------------------------------------------------------------
extern "C" void kernel_launch(void* const* d_in, const int* in_sizes, int n_in,
                              void* d_out, int out_size, void* d_ws,
                              size_t ws_size, hipStream_t stream) {
  (void)in_sizes; (void)n_in; (void)out_size; (void)ws_size;
  const float* hidden = (const float*)d_in[0];
  const float* mask   = (const float*)d_in[1];
  const float* cosb   = (const float*)d_in[2];
  const float* sinb   = (const float*)d_in[3];
  const float* kcache = (const float*)d_in[4];
  const float* vcache = (const float*)d_in[5];
  const int*   cbi    = (const int*)d_in[6];
  const int*   csi    = (const int*)d_in[7];
  const float* Wq     = (const float*)d_in[8];
  const float* Wk     = (const float*)d_in[9];
  const float* Wv     = (const float*)d_in[10];
  const float* Wo     = (const float*)d_in[11];
  const float* qnw    = (const float*)d_in[12];
  const float* knw    = (const float*)d_in[13];
  float* out = (float*)d_out;
  float* ws  = (float*)d_ws;

  // Workspace layout (floats)
  float* q   = ws;                                  // B*Q*NH*D  = 524288
  float* k   = q + (size_t)BB * QQ * NHH * DD;      // B*Q*KVH*D = 131072
  float* v   = k + (size_t)BB * QQ * KVHH * DD;     //           = 131072
  float* ctx = v + (size_t)BB * QQ * KVHH * DD;     // B*Q*NH*D  = 524288
  float* pO  = ctx + (size_t)BB * QQ * NHH * DD;    // 2048*Q*D  = 4194304
  float* pM  = pO + (size_t)BB * KVHH * GG * NCHUNK * QQ * DD;  // 32768
  float* pL  = pM + (size_t)BB * KVHH * GG * NCHUNK * QQ;       // 32768
  int*   map = (int*)(pL + (size_t)BB * KVHH * GG * NCHUNK * QQ);  // B*S ints

  // 1) QKV projections (weights streamed once through LDS, fp32 WMMA)
  gemm_rows128_kernel<<<dim3(NHH * DD / 16), 256, 0, stream>>>(hidden, Wq, q, HHID, NHH * DD);
  gemm_rows128_kernel<<<dim3(KVHH * DD / 16), 256, 0, stream>>>(hidden, Wk, k, HHID, KVHH * DD);
  gemm_rows128_kernel<<<dim3(KVHH * DD / 16), 256, 0, stream>>>(hidden, Wv, v, HHID, KVHH * DD);

  // 2) RMSNorm + RoPE on q and k (in place)
  rmsnorm_rope_kernel<<<dim3(BB * QQ * NHH / 8), 256, 0, stream>>>(q, qnw, cosb, sinb, NHH, BB * QQ * NHH);
  rmsnorm_rope_kernel<<<dim3(BB * QQ * KVHH / 8), 256, 0, stream>>>(k, knw, cosb, sinb, KVHH, BB * QQ * KVHH);

  // 3) Inverse scatter map for cache patch-on-read
  map_init_kernel<<<dim3((BB * SSEQ + 255) / 256), 256, 0, stream>>>(map, BB * SSEQ);
  map_scatter_kernel<<<dim3(1), 128, 0, stream>>>(map, cbi, csi, BB * QQ);

  // 4) Flash attention, split over S with online softmax partials
  attn_partial_kernel<<<dim3(BB * KVHH * GG * NCHUNK), 32, 0, stream>>>(
      q, k, v, kcache, vcache, mask, map, pO, pM, pL);
  attn_combine_kernel<<<dim3(BB * NHH * QQ), 128, 0, stream>>>(pO, pM, pL, ctx);

  // 5) Output projection -> d_out
  gemm_rows128_kernel<<<dim3(HHID / 16), 256, 0, stream>>>(ctx, Wo, out, NHH * DD, HHID);
}